// NeuralODE_80951543595559
// MI455X (gfx1250) — compile-verified
//
#include <hip/hip_runtime.h>
#include <cmath>

typedef float v2f __attribute__((ext_vector_type(2)));
typedef float v8f __attribute__((ext_vector_type(8)));

#define NW      8      // waves per workgroup (256 threads, wave32)
#define MROWS   32     // batch rows per workgroup
#define B_TOT   512
#define T_PTS   64
#define LAT     64
#define DHID    256

// ---- Tsit5 tableau (float), data-driven stage loop ----
__constant__ float ATAB[5][5] = {
    { 0.161f, 0.f, 0.f, 0.f, 0.f },
    { -0.008480655492356989f, 0.335480655492357f, 0.f, 0.f, 0.f },
    { 2.8971530571054935f, -6.359448489975075f, 4.3622954328695815f, 0.f, 0.f },
    { 5.325864828439257f, -11.748883564062828f, 7.4955393428898365f, -0.09249506636175525f, 0.f },
    { 5.86145544294642f, -12.92096931784711f, 8.159367898576159f, -0.071584973281401f, -0.028269050394068383f }
};
__constant__ float BTAB[6] = {
    0.09646076681806523f, 0.01f, 0.4798896504144996f,
    1.379008574103742f, -3.290069515436081f, 2.324710524099774f
};

__device__ __forceinline__ float softplus_f(float x) {
    // logaddexp(x, 0): branch-free, v_exp_f32 + v_log_f32
    return fmaxf(x, 0.0f) + __logf(1.0f + __expf(-fabsf(x)));
}

// out[M x N] = ACT?( in[M x K] @ W^T + bias ),  M = MROWS, W is [N x K] row-major.
// B-fragment B[k][n] = W[n][k] maps directly onto rows of W (v2f global loads).
// Fully-unrolled K loop of V_WMMA_F32_16X16X4_F32. 256 threads all active.
template <int K, int N, bool ACT>
__device__ __forceinline__ void dense_layer(const float* in_act,
                                            const float* __restrict__ W,
                                            const float* __restrict__ bias,
                                            float* out)
{
    constexpr int M = MROWS;
    constexpr int mtiles = M / 16;
    constexpr int ntiles = N / 16;
    const int lane   = threadIdx.x & 31;
    const int wv     = threadIdx.x >> 5;
    const int rc     = lane & 15;          // A row / B col within tile
    const int kb     = (lane >> 4) << 1;   // lanes 0-15 -> K 0,1 ; lanes 16-31 -> K 2,3
    const int rowoff = (lane >> 4) << 3;   // C/D: lanes 16-31 hold rows +8
    #pragma unroll 1
    for (int t = wv; t < mtiles * ntiles; t += NW) {
        const int mt = t % mtiles;
        const int nt = t / mtiles;
        const float* Ap = in_act + (mt * 16 + rc) * K + kb;
        const float* Wp = W      + (nt * 16 + rc) * K + kb;
        v8f acc = {0.f, 0.f, 0.f, 0.f, 0.f, 0.f, 0.f, 0.f};
        #pragma unroll
        for (int k0 = 0; k0 < K; k0 += 4) {
            v2f a = *(const v2f*)(Ap + k0);
            v2f b = *(const v2f*)(Wp + k0);
            acc = __builtin_amdgcn_wmma_f32_16x16x4_f32(
                      false, a, false, b, (short)0, acc, false, false);
        }
        const int n0 = nt * 16;
        const float bv = bias[n0 + rc];
        float* op = out + (mt * 16 + rowoff) * N + n0 + rc;
        #pragma unroll
        for (int r = 0; r < 8; ++r) {
            float v = acc[r] + bv;
            if (ACT) v = softplus_f(v);
            op[r * N] = v;
        }
    }
    __syncthreads();
}

// ---------------- encoder: x0 -> (mu, logvar) ----------------
__global__ __launch_bounds__(256)
void enc_kernel(const float* __restrict__ x0s,
                const float* __restrict__ W0, const float* __restrict__ W1,
                const float* __restrict__ W2, const float* __restrict__ W3,
                const float* __restrict__ b0, const float* __restrict__ b1,
                const float* __restrict__ b2, const float* __restrict__ b3,
                float* __restrict__ mu_out, float* __restrict__ lv_out)
{
    __shared__ float bufA[MROWS * 64];
    __shared__ float bufB[MROWS * 64];
    __shared__ float bufZ[MROWS * 128];
    const int m0  = blockIdx.x * MROWS;
    const int tid = threadIdx.x;
    for (int i = tid; i < MROWS * 16; i += 256)
        ((float4*)bufA)[i] = ((const float4*)(x0s + m0 * 64))[i];
    __syncthreads();
    dense_layer<64, 64, true >(bufA, W0, b0, bufB);
    dense_layer<64, 64, true >(bufB, W1, b1, bufA);
    dense_layer<64, 64, true >(bufA, W2, b2, bufB);
    dense_layer<64, 128, false>(bufB, W3, b3, bufZ);
    for (int i = tid; i < MROWS * 16; i += 256) {
        int r = i >> 4, c = i & 15;
        ((float4*)(mu_out + (m0 + r) * 64))[c] = ((const float4*)(bufZ + r * 128))[c];
        ((float4*)(lv_out + (m0 + r) * 64))[c] = ((const float4*)(bufZ + r * 128 + 64))[c];
    }
}

// ---------------- generator MLP: f(y) ----------------
__device__ __forceinline__ void gen_mlp(const float* in, float* out,
                                        const float* W0, const float* W1,
                                        const float* W2, const float* W3,
                                        const float* b0, const float* b1,
                                        const float* b2, const float* b3,
                                        float* hA, float* hB)
{
    dense_layer<64,   DHID, true >(in, W0, b0, hA);
    dense_layer<DHID, DHID, true >(hA, W1, b1, hB);
    dense_layer<DHID, DHID, true >(hB, W2, b2, hA);
    dense_layer<DHID, 64,   false>(hA, W3, b3, out);
}

// ---------------- persistent ODE integrator ----------------
__global__ __launch_bounds__(256)
void ode_kernel(const float* __restrict__ t_eval,
                const float* __restrict__ W0, const float* __restrict__ W1,
                const float* __restrict__ W2, const float* __restrict__ W3,
                const float* __restrict__ b0, const float* __restrict__ b1,
                const float* __restrict__ b2, const float* __restrict__ b3,
                const float* __restrict__ z0mu, float* __restrict__ zs_out)
{
    extern __shared__ float smem[];
    constexpr int NY  = MROWS * 64;
    constexpr int NY4 = NY / 4;
    float* y  = smem;             // [32 x 64] ODE state
    float* yt = y  + NY;          // [32 x 64] stage input
    float* ks = yt + NY;          // 6 x [32 x 64] stage derivatives
    float* hA = ks + 6 * NY;      // [32 x 256] MLP hidden
    float* hB = hA + MROWS * DHID;

    const int m0  = blockIdx.x * MROWS;
    const int tid = threadIdx.x;
    float4* y4  = (float4*)y;
    float4* yt4 = (float4*)yt;

    for (int i = tid; i < NY4; i += 256)
        y4[i] = ((const float4*)(z0mu + m0 * 64))[i];
    __syncthreads();
    // zs[:, t=0, :] = z0_mu
    for (int i = tid; i < NY4; i += 256) {
        int r = i >> 4, c = i & 15;
        ((float4*)(zs_out + ((size_t)(m0 + r) * T_PTS) * 64))[c] = y4[i];
    }

    for (int t = 0; t < T_PTS - 1; ++t) {
        const float h = (t_eval[t + 1] - t_eval[t]) * 0.5f;  // dt / N_SUB
        for (int sub = 0; sub < 2; ++sub) {
            #pragma unroll 1
            for (int s = 0; s < 6; ++s) {
                const float* in = (s == 0) ? y : yt;
                gen_mlp(in, ks + s * NY, W0, W1, W2, W3, b0, b1, b2, b3, hA, hB);
                if (s < 5) {
                    // yt = y + h * sum_{j<=s} A[s][j] * k_j
                    for (int i = tid; i < NY4; i += 256) {
                        float4 acc = {0.f, 0.f, 0.f, 0.f};
                        for (int j = 0; j <= s; ++j) {
                            const float aj = ATAB[s][j];
                            float4 kv = ((const float4*)(ks + j * NY))[i];
                            acc.x = fmaf(aj, kv.x, acc.x);
                            acc.y = fmaf(aj, kv.y, acc.y);
                            acc.z = fmaf(aj, kv.z, acc.z);
                            acc.w = fmaf(aj, kv.w, acc.w);
                        }
                        float4 yv = y4[i];
                        yv.x = fmaf(h, acc.x, yv.x);
                        yv.y = fmaf(h, acc.y, yv.y);
                        yv.z = fmaf(h, acc.z, yv.z);
                        yv.w = fmaf(h, acc.w, yv.w);
                        yt4[i] = yv;
                    }
                } else {
                    // y += h * sum_j B[j] * k_j
                    for (int i = tid; i < NY4; i += 256) {
                        float4 acc = {0.f, 0.f, 0.f, 0.f};
                        for (int j = 0; j < 6; ++j) {
                            const float bj = BTAB[j];
                            float4 kv = ((const float4*)(ks + j * NY))[i];
                            acc.x = fmaf(bj, kv.x, acc.x);
                            acc.y = fmaf(bj, kv.y, acc.y);
                            acc.z = fmaf(bj, kv.z, acc.z);
                            acc.w = fmaf(bj, kv.w, acc.w);
                        }
                        float4 yv = y4[i];
                        yv.x = fmaf(h, acc.x, yv.x);
                        yv.y = fmaf(h, acc.y, yv.y);
                        yv.z = fmaf(h, acc.z, yv.z);
                        yv.w = fmaf(h, acc.w, yv.w);
                        y4[i] = yv;
                    }
                }
                __syncthreads();
            }
        }
        for (int i = tid; i < NY4; i += 256) {
            int r = i >> 4, c = i & 15;
            ((float4*)(zs_out + ((size_t)(m0 + r) * T_PTS + (t + 1)) * 64))[c] = y4[i];
        }
    }
}

// ---------------- decoder GEMM over all B*T rows ----------------
__global__ __launch_bounds__(256)
void dec_kernel(const float* __restrict__ zs,
                const float* __restrict__ W, const float* __restrict__ b,
                float* __restrict__ xr)
{
    __shared__ float inb[MROWS * 64];
    __shared__ float outb[MROWS * 64];
    const int r0 = blockIdx.x * MROWS;   // flattened row index = b*T + t
    for (int i = threadIdx.x; i < MROWS * 16; i += 256)
        ((float4*)inb)[i] = ((const float4*)(zs + (size_t)r0 * 64))[i];
    __syncthreads();
    dense_layer<64, 64, false>(inb, W, b, outb);
    for (int i = threadIdx.x; i < MROWS * 16; i += 256)
        ((float4*)(xr + (size_t)r0 * 64))[i] = ((const float4*)outb)[i];
}

extern "C" void kernel_launch(void* const* d_in, const int* in_sizes, int n_in,
                              void* d_out, int out_size, void* d_ws, size_t ws_size,
                              hipStream_t stream)
{
    (void)in_sizes; (void)n_in; (void)out_size; (void)d_ws; (void)ws_size;
    const float* x0s = (const float*)d_in[0];
    const float* te  = (const float*)d_in[1];
    const float* eW0 = (const float*)d_in[2];
    const float* eW1 = (const float*)d_in[3];
    const float* eW2 = (const float*)d_in[4];
    const float* eW3 = (const float*)d_in[5];
    const float* eb0 = (const float*)d_in[6];
    const float* eb1 = (const float*)d_in[7];
    const float* eb2 = (const float*)d_in[8];
    const float* eb3 = (const float*)d_in[9];
    const float* gW0 = (const float*)d_in[10];
    const float* gW1 = (const float*)d_in[11];
    const float* gW2 = (const float*)d_in[12];
    const float* gW3 = (const float*)d_in[13];
    const float* gb0 = (const float*)d_in[14];
    const float* gb1 = (const float*)d_in[15];
    const float* gb2 = (const float*)d_in[16];
    const float* gb3 = (const float*)d_in[17];
    const float* drW = (const float*)d_in[18];
    const float* drb = (const float*)d_in[19];

    float* out = (float*)d_out;
    float* xr  = out;                                   // [B, T, DATA]
    float* zs  = out + (size_t)B_TOT * T_PTS * 64;      // [B, T, LAT]
    float* mu  = zs  + (size_t)B_TOT * T_PTS * 64;      // [B, LAT]
    float* lv  = mu  + (size_t)B_TOT * LAT;             // [B, LAT]

    enc_kernel<<<B_TOT / MROWS, 256, 0, stream>>>(
        x0s, eW0, eW1, eW2, eW3, eb0, eb1, eb2, eb3, mu, lv);

    const size_t odeShm = (size_t)(MROWS * 64 * 8 + MROWS * DHID * 2) * sizeof(float); // 128 KB
    ode_kernel<<<B_TOT / MROWS, 256, odeShm, stream>>>(
        te, gW0, gW1, gW2, gW3, gb0, gb1, gb2, gb3, mu, zs);

    dec_kernel<<<(B_TOT * T_PTS) / MROWS, 256, 0, stream>>>(zs, drW, drb, xr);
}